// SSDBoxHead_9466107920432
// MI455X (gfx1250) — compile-verified
//
#include <hip/hip_runtime.h>
#include <stdint.h>

// ---------------- problem constants (match reference) ----------------
#define B_          32
#define N_          24564
#define C_          81
#define NCLS        80           // foreground classes
#define PRE_NMS_    200
#define MAX_DET_    100
#define CAND_       512          // candidate list capacity for selection
#define SCORE_THR_  0.01f
#define NMS_THR_    0.45f
#define CENTER_VAR_ 0.1f
#define SIZE_VAR_   0.2f
#define BINS_       8192         // float-bit histogram bins (bits >> 17)

// =====================================================================
// Kernel 1: fused softmax statistics (row max, 1/sum(exp)) per (b,n).
// Streams the 255 MB logits tensor once through LDS using the gfx1250
// async global->LDS DMA path (ASYNCcnt), then row-reduces from LDS.
// =====================================================================
#define K1_ROWS 128
#define K1_DW   (K1_ROWS * C_)   // 10368 dwords = 41472 B per tile
#define K1_V4   (K1_DW / 4)      // 2592 b128 transfers

__global__ __launch_bounds__(K1_ROWS)
void softmax_stats_kernel(const float* __restrict__ logits,
                          float* __restrict__ maxv,
                          float* __restrict__ invs)
{
    __shared__ __align__(16) float tile[K1_DW];
    const int tid = threadIdx.x;
    const size_t blockRow = (size_t)blockIdx.x * K1_ROWS;
    const char* gbase = (const char*)(logits + blockRow * C_);
    const uint32_t lds_base = (uint32_t)(uintptr_t)tile;

    // cooperative async DMA of the whole tile into LDS (per-lane b128)
    for (int i = tid; i < K1_V4; i += K1_ROWS) {
        uint32_t loff = lds_base + (uint32_t)i * 16u;
        const char* gaddr = gbase + (size_t)i * 16;
        asm volatile("global_load_async_to_lds_b128 %0, %1, off"
                     :: "v"(loff), "v"(gaddr) : "memory");
    }
    asm volatile("s_wait_asynccnt 0" ::: "memory");
    __syncthreads();

    // one thread per row; stride 81 (odd) -> conflict-free LDS banking
    const float* row = &tile[tid * C_];
    float m = row[0];
    #pragma unroll 8
    for (int c = 1; c < C_; ++c) m = fmaxf(m, row[c]);
    float s = 0.f;
    #pragma unroll 8
    for (int c = 0; c < C_; ++c) s += __expf(row[c] - m);

    const size_t r = blockRow + tid;
    maxv[r] = m;
    invs[r] = 1.0f / s;
}

// =====================================================================
// Kernel 2: per-(image,class) top-200 selection + greedy NMS.
// grid = (80 classes, 32 images), 256 threads.
// =====================================================================
#define K2_T 256

__global__ __launch_bounds__(K2_T)
void per_class_nms_kernel(const float* __restrict__ logits,
                          const float* __restrict__ bbox,
                          const float* __restrict__ priors,
                          const float* __restrict__ maxv,
                          const float* __restrict__ invs,
                          float* __restrict__ cls_sc,
                          float* __restrict__ cls_bx)
{
    __shared__ int   hist[BINS_];
    __shared__ int   chunkSum[K2_T];
    __shared__ float candS[CAND_];
    __shared__ int   candI[CAND_];
    __shared__ int   sh_cut, sh_cnt;
    __shared__ float scS[PRE_NMS_];
    __shared__ float bxS[PRE_NMS_][4];
    __shared__ int   supp[PRE_NMS_];

    const int tid = threadIdx.x;
    const int cls = blockIdx.x;        // 0..79
    const int b   = blockIdx.y;        // 0..31
    const int c   = cls + 1;           // skip background
    const size_t rowBase = (size_t)b * N_;

    for (int i = tid; i < BINS_; i += K2_T) hist[i] = 0;
    __syncthreads();

    // pass 1: histogram of scores above threshold (float bits are monotonic)
    for (int n = tid; n < N_; n += K2_T) {
        const size_t r = rowBase + n;
        float p = __expf(logits[r * C_ + c] - maxv[r]) * invs[r];
        if (p > SCORE_THR_) {
            uint32_t bin = __float_as_uint(p) >> 17;
            if (bin > BINS_ - 1) bin = BINS_ - 1;
            atomicAdd(&hist[bin], 1);
        }
    }
    __syncthreads();

    // suffix-sum cutoff: largest bin such that count(bin >= cut) >= 200
    {
        int s = 0;
        const int base = tid * (BINS_ / K2_T);
        #pragma unroll 8
        for (int i = 0; i < BINS_ / K2_T; ++i) s += hist[base + i];
        chunkSum[tid] = s;
    }
    __syncthreads();
    if (tid == 0) {
        int acc = 0, chunk;
        for (chunk = K2_T - 1; chunk >= 0; --chunk) {
            if (acc + chunkSum[chunk] >= PRE_NMS_) break;
            acc += chunkSum[chunk];
        }
        int cb = 0;
        if (chunk >= 0) {
            const int lo = chunk * (BINS_ / K2_T);
            cb = lo;
            for (int bi = lo + (BINS_ / K2_T) - 1; bi >= lo; --bi) {
                if (acc + hist[bi] >= PRE_NMS_) { cb = bi; break; }
                acc += hist[bi];
            }
        }
        sh_cut = cb;
        sh_cnt = 0;
    }
    __syncthreads();
    const int cut = sh_cut;

    // pass 2: gather candidates with bin >= cut
    for (int n = tid; n < N_; n += K2_T) {
        const size_t r = rowBase + n;
        float p = __expf(logits[r * C_ + c] - maxv[r]) * invs[r];
        if (p > SCORE_THR_) {
            uint32_t bin = __float_as_uint(p) >> 17;
            if (bin > BINS_ - 1) bin = BINS_ - 1;
            if ((int)bin >= cut) {
                int pos = atomicAdd(&sh_cnt, 1);
                if (pos < CAND_) { candS[pos] = p; candI[pos] = n; }
            }
        }
    }
    __syncthreads();
    {
        const int cnt = (sh_cnt < CAND_) ? sh_cnt : CAND_;
        for (int i = tid; i < CAND_; i += K2_T)
            if (i >= cnt) { candS[i] = -1.0f; candI[i] = 0x7FFFFFFF; }
    }

    // bitonic sort (descending score, ties -> lower index). 256 pairs/substage.
    for (int k = 2; k <= CAND_; k <<= 1) {
        for (int j = k >> 1; j > 0; j >>= 1) {
            __syncthreads();
            const int i = 2 * j * (tid / j) + (tid % j);
            const int l = i + j;
            const bool up = ((i & k) == 0);
            float si = candS[i], sl = candS[l];
            int   ii = candI[i], il = candI[l];
            const bool iBetter = (si > sl) || ((si == sl) && (ii < il));
            if (up != iBetter) {
                candS[i] = sl; candS[l] = si;
                candI[i] = il; candI[l] = ii;
            }
        }
    }
    __syncthreads();

    // top-200: decode boxes for selected indices only
    for (int i = tid; i < PRE_NMS_; i += K2_T) {
        float s = candS[i];
        int   n = candI[i];
        if (s < SCORE_THR_ || n >= N_) { s = 0.f; n = 0; }
        scS[i]  = s;
        supp[i] = 0;
        const float px = priors[n * 4 + 0], py = priors[n * 4 + 1];
        const float pw = priors[n * 4 + 2], ph = priors[n * 4 + 3];
        const size_t off = (rowBase + (size_t)n) * 4;
        const float l0 = bbox[off + 0], l1 = bbox[off + 1];
        const float l2 = bbox[off + 2], l3 = bbox[off + 3];
        const float cx = l0 * CENTER_VAR_ * pw + px;
        const float cy = l1 * CENTER_VAR_ * ph + py;
        const float w  = __expf(l2 * SIZE_VAR_) * pw;
        const float h  = __expf(l3 * SIZE_VAR_) * ph;
        bxS[i][0] = cx - 0.5f * w; bxS[i][1] = cy - 0.5f * h;
        bxS[i][2] = cx + 0.5f * w; bxS[i][3] = cy + 0.5f * h;
    }
    __syncthreads();

    // greedy NMS (sequential over i, parallel over j)
    for (int i = 0; i < PRE_NMS_; ++i) {
        const bool active = (scS[i] > 0.f) && (supp[i] == 0);
        if (active) {
            const float x1 = bxS[i][0], y1 = bxS[i][1];
            const float x2 = bxS[i][2], y2 = bxS[i][3];
            const float ai = fmaxf(x2 - x1, 0.f) * fmaxf(y2 - y1, 0.f);
            for (int jx = tid; jx < PRE_NMS_; jx += K2_T) {
                if (jx > i) {
                    const float u1 = fmaxf(x1, bxS[jx][0]);
                    const float v1 = fmaxf(y1, bxS[jx][1]);
                    const float u2 = fminf(x2, bxS[jx][2]);
                    const float v2 = fminf(y2, bxS[jx][3]);
                    const float inter = fmaxf(u2 - u1, 0.f) * fmaxf(v2 - v1, 0.f);
                    const float aj = fmaxf(bxS[jx][2] - bxS[jx][0], 0.f) *
                                     fmaxf(bxS[jx][3] - bxS[jx][1], 0.f);
                    const float iou = inter / fmaxf(ai + aj - inter, 1e-9f);
                    if (iou > NMS_THR_) supp[jx] = 1;
                }
            }
        }
        __syncthreads();
    }

    // emit kept scores + boxes to workspace
    const size_t obase = ((size_t)b * NCLS + cls) * PRE_NMS_;
    for (int i = tid; i < PRE_NMS_; i += K2_T) {
        const bool keep = (scS[i] > 0.f) && (supp[i] == 0);
        cls_sc[obase + i] = keep ? scS[i] : 0.f;
        cls_bx[(obase + i) * 4 + 0] = bxS[i][0];
        cls_bx[(obase + i) * 4 + 1] = bxS[i][1];
        cls_bx[(obase + i) * 4 + 2] = bxS[i][2];
        cls_bx[(obase + i) * 4 + 3] = bxS[i][3];
    }
}

// =====================================================================
// Kernel 3: per-image final top-100 over 80*200 kept scores.
// grid = 32 images, 256 threads.
// =====================================================================
#define K3_T 256

__global__ __launch_bounds__(K3_T)
void topk_final_kernel(const float* __restrict__ cls_sc,
                       const float* __restrict__ cls_bx,
                       float* __restrict__ out_bx,
                       float* __restrict__ out_sc,
                       int*   __restrict__ out_lb)
{
    __shared__ int   hist[BINS_];
    __shared__ int   chunkSum[K3_T];
    __shared__ float candS[CAND_];
    __shared__ int   candI[CAND_];
    __shared__ int   sh_cut, sh_cnt;

    const int tid  = threadIdx.x;
    const int b    = blockIdx.x;
    const int FLAT = NCLS * PRE_NMS_;   // 16000
    const float* sc = cls_sc + (size_t)b * FLAT;

    for (int i = tid; i < BINS_; i += K3_T) hist[i] = 0;
    __syncthreads();

    for (int n = tid; n < FLAT; n += K3_T) {
        const float p = sc[n];
        if (p > 0.f) {
            uint32_t bin = __float_as_uint(p) >> 17;
            if (bin > BINS_ - 1) bin = BINS_ - 1;
            atomicAdd(&hist[bin], 1);
        }
    }
    __syncthreads();
    {
        int s = 0;
        const int base = tid * (BINS_ / K3_T);
        #pragma unroll 8
        for (int i = 0; i < BINS_ / K3_T; ++i) s += hist[base + i];
        chunkSum[tid] = s;
    }
    __syncthreads();
    if (tid == 0) {
        int acc = 0, chunk;
        for (chunk = K3_T - 1; chunk >= 0; --chunk) {
            if (acc + chunkSum[chunk] >= MAX_DET_) break;
            acc += chunkSum[chunk];
        }
        int cb = 0;
        if (chunk >= 0) {
            const int lo = chunk * (BINS_ / K3_T);
            cb = lo;
            for (int bi = lo + (BINS_ / K3_T) - 1; bi >= lo; --bi) {
                if (acc + hist[bi] >= MAX_DET_) { cb = bi; break; }
                acc += hist[bi];
            }
        }
        sh_cut = cb;
        sh_cnt = 0;
    }
    __syncthreads();
    const int cut = sh_cut;

    for (int n = tid; n < FLAT; n += K3_T) {
        const float p = sc[n];
        if (p > 0.f) {
            uint32_t bin = __float_as_uint(p) >> 17;
            if (bin > BINS_ - 1) bin = BINS_ - 1;
            if ((int)bin >= cut) {
                int pos = atomicAdd(&sh_cnt, 1);
                if (pos < CAND_) { candS[pos] = p; candI[pos] = n; }
            }
        }
    }
    __syncthreads();
    {
        const int cnt = (sh_cnt < CAND_) ? sh_cnt : CAND_;
        for (int i = tid; i < CAND_; i += K3_T)
            if (i >= cnt) { candS[i] = -1.0f; candI[i] = 0x7FFFFFFF; }
    }

    for (int k = 2; k <= CAND_; k <<= 1) {
        for (int j = k >> 1; j > 0; j >>= 1) {
            __syncthreads();
            const int i = 2 * j * (tid / j) + (tid % j);
            const int l = i + j;
            const bool up = ((i & k) == 0);
            float si = candS[i], sl = candS[l];
            int   ii = candI[i], il = candI[l];
            const bool iBetter = (si > sl) || ((si == sl) && (ii < il));
            if (up != iBetter) {
                candS[i] = sl; candS[l] = si;
                candI[i] = il; candI[l] = ii;
            }
        }
    }
    __syncthreads();

    for (int i = tid; i < MAX_DET_; i += K3_T) {
        float s = candS[i];
        int   f = candI[i];
        if (s <= 0.f || f >= FLAT) { s = 0.f; f = 0; }
        const size_t o = (size_t)b * MAX_DET_ + i;
        out_sc[o] = s;
        out_lb[o] = f / PRE_NMS_ + 1;
        const float* bp = cls_bx + ((size_t)b * FLAT + f) * 4;
        out_bx[o * 4 + 0] = bp[0];
        out_bx[o * 4 + 1] = bp[1];
        out_bx[o * 4 + 2] = bp[2];
        out_bx[o * 4 + 3] = bp[3];
    }
}

// =====================================================================
// launch
// =====================================================================
extern "C" void kernel_launch(void* const* d_in, const int* in_sizes, int n_in,
                              void* d_out, int out_size, void* d_ws, size_t ws_size,
                              hipStream_t stream)
{
    const float* logits = (const float*)d_in[0];   // (32, 24564, 81)
    const float* bbox   = (const float*)d_in[1];   // (32, 24564, 4)
    const float* priors = (const float*)d_in[2];   // (24564, 4)

    float* ws     = (float*)d_ws;
    float* maxv   = ws;                                         // B*N
    float* invs   = maxv + (size_t)B_ * N_;                     // B*N
    float* cls_sc = invs + (size_t)B_ * N_;                     // B*80*200
    float* cls_bx = cls_sc + (size_t)B_ * NCLS * PRE_NMS_;      // B*80*200*4

    float* out_bx = (float*)d_out;                              // B*100*4
    float* out_sc = out_bx + (size_t)B_ * MAX_DET_ * 4;         // B*100
    int*   out_lb = (int*)(out_sc + (size_t)B_ * MAX_DET_);     // B*100

    softmax_stats_kernel<<<(B_ * N_) / K1_ROWS, K1_ROWS, 0, stream>>>(logits, maxv, invs);

    dim3 g2(NCLS, B_);
    per_class_nms_kernel<<<g2, K2_T, 0, stream>>>(logits, bbox, priors, maxv, invs,
                                                  cls_sc, cls_bx);

    topk_final_kernel<<<B_, K3_T, 0, stream>>>(cls_sc, cls_bx, out_bx, out_sc, out_lb);
}